// SineActivation_59742995087547
// MI455X (gfx1250) — compile-verified
//
#include <hip/hip_runtime.h>

// Time2Vec / SineActivation for MI455X (gfx1250, wave32).
//
// out[row, j] = sin(tau[row]*w[j] + b[j])  for j < 255
// out[row, 255] = tau[row]*w0 + b0
//
// Store-bandwidth bound (268 MB out @ 23.3 TB/s ~= 11.5 us). The affine part
// is computed with V_WMMA_F32_16X16X4_F32 (full f32 precision rank-1 A*B+C),
// sin via hardware v_sin_f32, stores are non-temporal (output > L2 capacity).

typedef __attribute__((ext_vector_type(2))) float v2f;
typedef __attribute__((ext_vector_type(8))) float v8f;

#define OUTF 256
#define WAVES_PER_BLOCK 8

__global__ __launch_bounds__(256) void t2v_wmma_kernel(
    const float* __restrict__ tau,
    const float* __restrict__ w,    // [255]
    const float* __restrict__ b,    // [255]
    const float* __restrict__ w0,   // [1]
    const float* __restrict__ b0,   // [1]
    float* __restrict__ out,        // [nRows, 256]
    int nRows)
{
    __shared__ float s_w[OUTF];
    __shared__ float s_b[OUTF];

    const int tid = threadIdx.x;
    if (tid < OUTF - 1) {
        s_w[tid] = w[tid];
        s_b[tid] = b[tid];
    } else if (tid == OUTF - 1) {
        s_w[OUTF - 1] = w0[0];   // linear column patched into the tables
        s_b[OUTF - 1] = b0[0];
    }
    __syncthreads();

    const int lane = tid & 31;
    const int wave = tid >> 5;
    const int rowTile = blockIdx.x * WAVES_PER_BLOCK + wave;
    const int row0 = rowTile * 16;
    if (row0 + 16 > nRows) return;          // wave-uniform guard (after barrier)

    const int  n  = lane & 15;              // N index (column-in-tile) / M for A
    const bool lo = lane < 16;

    // A-matrix (16x4 f32): lanes 0-15, VGPR0 = A[m][K=0] = tau[row0+m].
    // All other (lane,VGPR) slots are K>0 halves -> zero them.
    const float t = tau[row0 + n];          // lanes 16-31 load dupes (cached)
    v2f A;
    A.x = lo ? t : 0.0f;
    A.y = 0.0f;

    // Per-lane output base: VGPR r of D holds row row0+r (lanes 0-15) or
    // row0+r+8 (lanes 16-31), column c+n.
    float* outLane = out + (size_t)(row0 + (lo ? 0 : 8)) * OUTF + n;

    for (int ct = 0; ct < 16; ++ct) {
        const int c = ct * 16;
        const float wv = s_w[c + n];
        const float bv = s_b[c + n];

        // B-matrix (4x16 f32): row K=0 = w[c..c+15] in lanes 0-15 of VGPR0;
        // everything else zero (only K=0 contributes to the rank-1 product).
        v2f B;
        B.x = lo ? wv : 0.0f;
        B.y = 0.0f;

        // C-matrix (16x16 f32): C[m][n] = b[c+n] -> every C VGPR = bv.
        v8f C;
#pragma unroll
        for (int i = 0; i < 8; ++i) C[i] = bv;

        // D = A x B + C  ->  D[m][n] = tau[row0+m]*w[c+n] + b[c+n]
        v8f D = __builtin_amdgcn_wmma_f32_16x16x4_f32(
            /*neg_a=*/false, A, /*neg_b=*/false, B,
            /*c_mod=*/(short)0, C, /*reuse_a=*/false, /*reuse_b=*/false);

        // Column 255 (last tile, n==15) is the linear term: no sin.
        const bool linearCol = (ct == 15) && (n == 15);

#pragma unroll
        for (int r = 0; r < 8; ++r) {
            float v = D[r];
            // v_sin_f32 computes sin(2*pi*x): pre-scale by 1/(2*pi).
            float sv = __builtin_amdgcn_sinf(v * 0.15915494309189535f);
            v = linearCol ? v : sv;
            __builtin_nontemporal_store(v, outLane + (size_t)r * OUTF + c);
        }
    }
}

extern "C" void kernel_launch(void* const* d_in, const int* in_sizes, int n_in,
                              void* d_out, int out_size, void* d_ws, size_t ws_size,
                              hipStream_t stream) {
    const float* tau = (const float*)d_in[0];   // [B*S]
    const float* w   = (const float*)d_in[1];   // [255]
    const float* b   = (const float*)d_in[2];   // [255]
    const float* w0  = (const float*)d_in[3];   // [1]
    const float* b0  = (const float*)d_in[4];   // [1]
    float* out = (float*)d_out;

    const int nRows = in_sizes[0];                       // 262144
    const int rowTiles = nRows / 16;                     // 16 rows per wave
    const int blocks = (rowTiles + WAVES_PER_BLOCK - 1) / WAVES_PER_BLOCK;

    t2v_wmma_kernel<<<blocks, 256, 0, stream>>>(tau, w, b, w0, b0, out, nRows);
}